// VariationalQuantumEigensolver_14877766713538
// MI455X (gfx1250) — compile-verified
//
#include <hip/hip_runtime.h>
#include <math.h>

// ---------------------------------------------------------------------------
// VQE collapses to:  state[i] = e^{i*sum(params)/2}/sqrt(8192)  (scalar phase),
//                    energy   = sum(H) / 8192                   (256MB reduce).
// Reduction is HBM-bound (~11.5us @ 23.3TB/s); accumulate via fp32 WMMA
// (B = ones) so the matrix pipe does the adds: D = A*1 + C absorbs 64 f32
// per v_wmma_f32_16x16x4_f32.
// ---------------------------------------------------------------------------

typedef __attribute__((ext_vector_type(2))) float v2f;
typedef __attribute__((ext_vector_type(8))) float v8f;

#define STATE_DIM   8192
#define NPARAMS     104
#define RED_BLOCKS  2048
#define RED_THREADS 256
// H has 8192*8192 = 16,777,216 float4's; / (2048*256) threads = 32 per thread.
#define VEC4_PER_THREAD 32

__global__ __launch_bounds__(256)
void vqe_state_kernel(const float* __restrict__ params, float* __restrict__ out) {
    // Faithful emulation of the scan: z *= e^{i*angle/2}; z /= |z|.
    float zr = 1.0f, zi = 0.0f;
    for (int k = 0; k < NPARAMS; ++k) {
        float h = params[k] * 0.5f;
        float c = cosf(h), s = sinf(h);
        float nr = zr * c - zi * s;
        float ni = zr * s + zi * c;
        float inv = rsqrtf(nr * nr + ni * ni);
        zr = nr * inv;
        zi = ni * inv;
    }
    const float amp = 0.011048543456039806f; // 1/sqrt(8192)
    int idx = blockIdx.x * blockDim.x + threadIdx.x;
    if (idx < STATE_DIM) {
        out[1 + 2 * idx]     = zr * amp;  // real
        out[1 + 2 * idx + 1] = zi * amp;  // imag
    }
}

__device__ __forceinline__ float wave32_reduce(float s) {
    #pragma unroll
    for (int off = 16; off > 0; off >>= 1)
        s += __shfl_xor(s, off, 32);
    return s;
}

__global__ __launch_bounds__(RED_THREADS)
void vqe_hsum_kernel(const float* __restrict__ H, float* __restrict__ partials) {
    const float4* __restrict__ Hv = (const float4*)H;
    // Contiguous 128KB segment per block, coalesced within block.
    size_t base = (size_t)blockIdx.x * (VEC4_PER_THREAD * RED_THREADS) + threadIdx.x;

    v8f acc = {};                 // 16x16 f32 accumulator (8 VGPRs/lane)
    v2f bones; bones[0] = 1.0f; bones[1] = 1.0f;   // B = ones (4x16)

    #pragma unroll
    for (int i = 0; i < VEC4_PER_THREAD; ++i) {
        float4 v = Hv[base + (size_t)i * RED_THREADS];
        if (i + 8 < VEC4_PER_THREAD)  // stays inside this block's segment
            __builtin_prefetch(&Hv[base + (size_t)(i + 8) * RED_THREADS], 0, 1);
        v2f a0; a0[0] = v.x; a0[1] = v.y;
        v2f a1; a1[0] = v.z; a1[1] = v.w;
        // D = A*ones + C : row-sums of A accumulate into every column of C.
        acc = __builtin_amdgcn_wmma_f32_16x16x4_f32(
                  false, a0, false, bones, (short)0, acc, false, false);
        acc = __builtin_amdgcn_wmma_f32_16x16x4_f32(
                  false, a1, false, bones, (short)0, acc, false, false);
    }

    float s = acc[0] + acc[1] + acc[2] + acc[3] +
              acc[4] + acc[5] + acc[6] + acc[7];
    s = wave32_reduce(s);

    __shared__ float wsum[RED_THREADS / 32];
    int lane = threadIdx.x & 31;
    int wid  = threadIdx.x >> 5;
    if (lane == 0) wsum[wid] = s;
    __syncthreads();
    if (threadIdx.x == 0) {
        float t = 0.0f;
        #pragma unroll
        for (int w = 0; w < RED_THREADS / 32; ++w) t += wsum[w];
        // Each element was counted 16x (16 D-columns).
        partials[blockIdx.x] = t * (1.0f / 16.0f);
    }
}

__global__ __launch_bounds__(256)
void vqe_final_kernel(const float* __restrict__ partials, float* __restrict__ out) {
    float s = 0.0f;
    for (int i = threadIdx.x; i < RED_BLOCKS; i += 256)
        s += partials[i];
    s = wave32_reduce(s);

    __shared__ float wsum[8];
    int lane = threadIdx.x & 31;
    int wid  = threadIdx.x >> 5;
    if (lane == 0) wsum[wid] = s;
    __syncthreads();
    if (threadIdx.x == 0) {
        float t = 0.0f;
        #pragma unroll
        for (int w = 0; w < 8; ++w) t += wsum[w];
        out[0] = t * (1.0f / 8192.0f);   // energy = sum(H)/STATE_DIM
    }
}

extern "C" void kernel_launch(void* const* d_in, const int* in_sizes, int n_in,
                              void* d_out, int out_size, void* d_ws, size_t ws_size,
                              hipStream_t stream) {
    const float* params = (const float*)d_in[0];  // 104 fp32
    const float* H      = (const float*)d_in[1];  // 8192*8192 fp32
    float* out      = (float*)d_out;              // [energy, re/im * 8192]
    float* partials = (float*)d_ws;               // RED_BLOCKS floats (8KB)

    vqe_state_kernel<<<STATE_DIM / 256, 256, 0, stream>>>(params, out);
    vqe_hsum_kernel<<<RED_BLOCKS, RED_THREADS, 0, stream>>>(H, partials);
    vqe_final_kernel<<<1, 256, 0, stream>>>(partials, out);
}